// EdgeClassifier_59648505806951
// MI455X (gfx1250) — compile-verified
//
#include <hip/hip_runtime.h>

typedef float v2f __attribute__((ext_vector_type(2)));
typedef float v4f __attribute__((ext_vector_type(4)));
typedef float v8f __attribute__((ext_vector_type(8)));

#define N_NODES 100000
#define N_EDGES 1600000
#define HID 128
#define N_MTILES (N_NODES / 16)   // 6250
#define E_MTILES (N_EDGES / 16)   // 100000

// ---------------------------------------------------------------- utilities
__global__ void zero_f32(float* __restrict__ p, int n) {
    int i = blockIdx.x * blockDim.x + threadIdx.x;
    int stride = gridDim.x * blockDim.x;
    for (; i < n; i += stride) p[i] = 0.0f;
}

__global__ void degree_kernel(const int* __restrict__ dst, int* __restrict__ deg) {
    int e = blockIdx.x * blockDim.x + threadIdx.x;
    if (e < N_EDGES) atomicAdd(&deg[dst[e]], 1);
}

__global__ void invdeg_kernel(const int* __restrict__ deg, float* __restrict__ invd) {
    int i = blockIdx.x * blockDim.x + threadIdx.x;
    if (i < N_NODES) invd[i] = 1.0f / fmaxf((float)deg[i], 1.0f);
}

// ------------------------------------------------- scatter: agg[dst] += h[src]
// one wave per edge; float4 coalesced row read, hardware f32 atomics into agg.
__global__ void __launch_bounds__(256)
scatter_kernel(const float* __restrict__ h, const int* __restrict__ src,
               const int* __restrict__ dst, float* __restrict__ agg) {
    int e    = blockIdx.x * (blockDim.x >> 5) + (threadIdx.x >> 5);
    int lane = threadIdx.x & 31;
    int s = src[e], d = dst[e];
    v4f v = *(const v4f*)(h + (size_t)s * HID + lane * 4);
    float* ap = agg + (size_t)d * HID + lane * 4;
    unsafeAtomicAdd(ap + 0, v.x);
    unsafeAtomicAdd(ap + 1, v.y);
    unsafeAtomicAdd(ap + 2, v.z);
    unsafeAtomicAdd(ap + 3, v.w);
}

// ---------------------------------------------------------------- SAGE GEMM
// out = leaky_relu( h @ ws + (agg * inv_deg) @ wn + b )
// One wave handles a 16-row M tile and the full N=128 (8 WMMA N-tiles).
// fp32 WMMA 16x16x4 layouts (ISA 7.12.2):
//   A: lane m = lane&15, VGPR{0,1} = K = (lane>>4)*2 + {0,1}
//   B: lane n = lane&15, VGPR{0,1} = K = (lane>>4)*2 + {0,1}
//   C: VGPR r -> row = r + (lane>>4)*8, col = lane&15
__global__ void __launch_bounds__(256)
sage_gemm_kernel(const float* __restrict__ h, const float* __restrict__ agg,
                 const float* __restrict__ invd,
                 const float* __restrict__ ws, const float* __restrict__ wn,
                 const float* __restrict__ bias, float* __restrict__ out) {
    int t = blockIdx.x * (blockDim.x >> 5) + (threadIdx.x >> 5);
    if (t >= N_MTILES) return;                 // wave-uniform guard (EXEC stays full)
    int lane = threadIdx.x & 31;
    int m    = lane & 15;
    int hi   = lane >> 4;                      // 0 or 1

    int row = t * 16 + m;
    const float* hrow = h   + (size_t)row * HID;
    const float* arow = agg + (size_t)row * HID;
    float idg = invd[row];

    v8f zero8 = {0.f, 0.f, 0.f, 0.f, 0.f, 0.f, 0.f, 0.f};
    v8f acc[8];
#pragma unroll
    for (int i = 0; i < 8; ++i) acc[i] = zero8;

    for (int k0 = 0; k0 < HID; k0 += 4) {
        int kb = k0 + hi * 2;
        v2f a_s = *(const v2f*)(hrow + kb);
        v2f a_n = *(const v2f*)(arow + kb);
        a_n *= idg;
#pragma unroll
        for (int nt = 0; nt < 8; ++nt) {
            int n = nt * 16 + m;
            v2f bs; bs.x = ws[kb * HID + n]; bs.y = ws[(kb + 1) * HID + n];
            v2f bn; bn.x = wn[kb * HID + n]; bn.y = wn[(kb + 1) * HID + n];
            acc[nt] = __builtin_amdgcn_wmma_f32_16x16x4_f32(
                false, a_s, false, bs, (short)0, acc[nt], false, false);
            acc[nt] = __builtin_amdgcn_wmma_f32_16x16x4_f32(
                false, a_n, false, bn, (short)0, acc[nt], false, false);
        }
    }

#pragma unroll
    for (int nt = 0; nt < 8; ++nt) {
        float bv = bias[nt * 16 + m];
#pragma unroll
        for (int r = 0; r < 8; ++r) {
            float x = acc[nt][r] + bv;
            x = (x > 0.0f) ? x : 0.01f * x;    // leaky_relu(0.01)
            int orow = t * 16 + r + hi * 8;
            out[(size_t)orow * HID + nt * 16 + m] = x;
        }
    }
}

// ---------------------------------------------------------------- edge MLP
// hidden = relu([h[src], h[dst], ef] @ mw1 + mb1)   (K = 128 + 128 + 1)
// logits = hidden @ mw2 + mb2                        (128 -> 2, shfl reduction)
__global__ void __launch_bounds__(256)
edge_mlp_kernel(const float* __restrict__ h, const float* __restrict__ ef,
                const int* __restrict__ src, const int* __restrict__ dst,
                const float* __restrict__ mw1, const float* __restrict__ mb1,
                const float* __restrict__ mw2, const float* __restrict__ mb2,
                float* __restrict__ out) {
    int t = blockIdx.x * (blockDim.x >> 5) + (threadIdx.x >> 5);
    if (t >= E_MTILES) return;
    int lane = threadIdx.x & 31;
    int m    = lane & 15;
    int hi   = lane >> 4;

    int e  = t * 16 + m;
    int ns = src[e], nd = dst[e];
    float efv = ef[e];                          // edge feature for edge m of tile
    const float* hs = h + (size_t)ns * HID;
    const float* hd = h + (size_t)nd * HID;

    v8f zero8 = {0.f, 0.f, 0.f, 0.f, 0.f, 0.f, 0.f, 0.f};
    v8f acc[8];
#pragma unroll
    for (int i = 0; i < 8; ++i) acc[i] = zero8;

    for (int k0 = 0; k0 < HID; k0 += 4) {
        int kb = k0 + hi * 2;
        v2f a0 = *(const v2f*)(hs + kb);        // K rows [0,128)  : src feats
        v2f a1 = *(const v2f*)(hd + kb);        // K rows [128,256): dst feats
#pragma unroll
        for (int nt = 0; nt < 8; ++nt) {
            int n = nt * 16 + m;
            v2f b0; b0.x = mw1[kb * HID + n];          b0.y = mw1[(kb + 1) * HID + n];
            v2f b1; b1.x = mw1[(128 + kb) * HID + n];  b1.y = mw1[(129 + kb) * HID + n];
            acc[nt] = __builtin_amdgcn_wmma_f32_16x16x4_f32(
                false, a0, false, b0, (short)0, acc[nt], false, false);
            acc[nt] = __builtin_amdgcn_wmma_f32_16x16x4_f32(
                false, a1, false, b1, (short)0, acc[nt], false, false);
        }
    }

    // K row 256 (edge feature) + bias + relu, all in registers
#pragma unroll
    for (int nt = 0; nt < 8; ++nt) {
        float wef = mw1[256 * HID + nt * 16 + m];
        float bv  = mb1[nt * 16 + m];
#pragma unroll
        for (int r = 0; r < 8; ++r) {
            float efr = __shfl(efv, r + hi * 8);   // ef of this C-row's edge
            float x = acc[nt][r] + efr * wef + bv;
            acc[nt][r] = fmaxf(x, 0.0f);
        }
    }

    // logits: 128 -> 2 via cross-lane reduction over the 16 lanes of each half
    float w20[8], w21[8];
#pragma unroll
    for (int nt = 0; nt < 8; ++nt) {
        w20[nt] = mw2[(nt * 16 + m) * 2 + 0];
        w21[nt] = mw2[(nt * 16 + m) * 2 + 1];
    }
    float c0 = mb2[0], c1 = mb2[1];
#pragma unroll
    for (int r = 0; r < 8; ++r) {
        float p0 = 0.f, p1 = 0.f;
#pragma unroll
        for (int nt = 0; nt < 8; ++nt) {
            p0 += acc[nt][r] * w20[nt];
            p1 += acc[nt][r] * w21[nt];
        }
        p0 += __shfl_xor(p0, 1); p0 += __shfl_xor(p0, 2);
        p0 += __shfl_xor(p0, 4); p0 += __shfl_xor(p0, 8);
        p1 += __shfl_xor(p1, 1); p1 += __shfl_xor(p1, 2);
        p1 += __shfl_xor(p1, 4); p1 += __shfl_xor(p1, 8);
        if (m == 0) {
            int orow = t * 16 + r + hi * 8;
            out[orow * 2 + 0] = p0 + c0;
            out[orow * 2 + 1] = p1 + c1;
        }
    }
}

// ---------------------------------------------------------------- launcher
extern "C" void kernel_launch(void* const* d_in, const int* in_sizes, int n_in,
                              void* d_out, int out_size, void* d_ws, size_t ws_size,
                              hipStream_t stream) {
    const float* node_feats = (const float*)d_in[0];
    const float* edge_feats = (const float*)d_in[1];
    const int*   src        = (const int*)d_in[2];
    const int*   dst        = (const int*)d_in[3];
    const float* wss[3] = {(const float*)d_in[4], (const float*)d_in[7], (const float*)d_in[10]};
    const float* wns[3] = {(const float*)d_in[5], (const float*)d_in[8], (const float*)d_in[11]};
    const float* bbs[3] = {(const float*)d_in[6], (const float*)d_in[9], (const float*)d_in[12]};
    const float* mw1 = (const float*)d_in[13];
    const float* mb1 = (const float*)d_in[14];
    const float* mw2 = (const float*)d_in[15];
    const float* mb2 = (const float*)d_in[16];

    char* base = (char*)d_ws;
    float* invd = (float*)base;                                          // 400 KB
    int*   deg  = (int*)(base + ((size_t)1 << 19));                      // @512 KB, 400 KB
    float* agg  = (float*)(base + ((size_t)1 << 20));                    // @1 MB, 51.2 MB
    float* hA   = (float*)(base + ((size_t)1 << 20) + ((size_t)52 << 20));  // 51.2 MB
    float* hB   = (float*)(base + ((size_t)1 << 20) + ((size_t)104 << 20)); // 51.2 MB

    // degrees -> inv_deg
    zero_f32<<<1024, 256, 0, stream>>>((float*)deg, N_NODES);
    degree_kernel<<<(N_EDGES + 255) / 256, 256, 0, stream>>>(dst, deg);
    invdeg_kernel<<<(N_NODES + 255) / 256, 256, 0, stream>>>(deg, invd);

    const float* hin = node_feats;
    float* houts[3] = {hA, hB, hA};
    for (int L = 0; L < 3; ++L) {
        zero_f32<<<2048, 256, 0, stream>>>(agg, N_NODES * HID);
        scatter_kernel<<<N_EDGES / 8, 256, 0, stream>>>(hin, src, dst, agg);
        sage_gemm_kernel<<<(N_MTILES + 7) / 8, 256, 0, stream>>>(
            hin, agg, invd, wss[L], wns[L], bbs[L], houts[L]);
        hin = houts[L];
    }

    edge_mlp_kernel<<<E_MTILES / 8, 256, 0, stream>>>(
        hin, edge_feats, src, dst, mw1, mb1, mw2, mb2, (float*)d_out);
}